// Net_84052509982900
// MI455X (gfx1250) — compile-verified
//
#include <hip/hip_runtime.h>
#include <cmath>

typedef float v2f __attribute__((ext_vector_type(2)));
typedef float v8f __attribute__((ext_vector_type(8)));

#define ACT_ELEMS 6291456   // 128*3*128*128 == 128*12*64*64 == ... (squeeze-invariant)
#define HALF_LOG_2PI 0.918938533204672742f

// ---------------------------------------------------------------------------
// init per-sample / scalar logdet accumulators
// ---------------------------------------------------------------------------
__global__ void init_accum(float* ld_ps, float* ld_sc, float c0) {
    int t = threadIdx.x;
    if (t < 128) ld_ps[t] = c0;     // d*(log(0.1)+log(0.99)) logit constant
    if (t == 128) *ld_sc = 0.f;
}

// ---------------------------------------------------------------------------
// weight transpose into WMMA-A friendly layout:
//   Kt[tap][ci0/4][kh][co][p] , ci = 4*(ci0/4) + 2*kh + p , co padded to CPAD
// Lane (m=co) then loads its two A values as one aligned 8-byte global load.
// ---------------------------------------------------------------------------
__global__ void transpose_w(const float* __restrict__ K, float* __restrict__ Kt,
                            int C, int CPAD) {
    int tot = 9 * C * CPAD;
    for (int i = blockIdx.x * blockDim.x + threadIdx.x; i < tot; i += gridDim.x * blockDim.x) {
        int p    = i & 1;
        int co   = (i >> 1) % CPAD;
        int r    = i / (2 * CPAD);
        int kh   = r & 1;
        int step = r >> 1;
        int c4s  = C / 4;
        int tap  = step / c4s;
        int c4   = step % c4s;
        int ci   = c4 * 4 + kh * 2 + p;
        Kt[i] = (co < C) ? K[(co * C + ci) * 9 + tap] : 0.f;
    }
}

// ---------------------------------------------------------------------------
// logit layer + per-sample logdet partial
// ---------------------------------------------------------------------------
__global__ void logit_fwd(const float* __restrict__ x, float* __restrict__ out,
                          float* __restrict__ ld_ps) {
    __shared__ float red[256];
    const int b = blockIdx.y;
    const int idx = blockIdx.x * 256 + threadIdx.x;
    const size_t base = (size_t)b * 49152;
    float xv = x[base + idx];
    float xs = 0.005f + 0.99f * xv;
    float l1 = logf(xs), l2 = logf(1.f - xs);
    out[base + idx] = 0.1f * (l1 - l2);
    red[threadIdx.x] = -(l1 + l2);
    __syncthreads();
    for (int s = 128; s > 0; s >>= 1) {
        if (threadIdx.x < s) red[threadIdx.x] += red[threadIdx.x + s];
        __syncthreads();
    }
    if (threadIdx.x == 0) atomicAdd(&ld_ps[b], red[0]);
}

// ---------------------------------------------------------------------------
// fused squeeze + actnorm + circular 3x3 conv (WMMA f32) + bias + gate/gauss
// One workgroup per (image, output row). MT*NT waves, each a 16x16 D tile.
// Depth-2 software pipeline on the flattened (tap, ci0) K loop.
// LDS staging is channel-pair interleaved: B fragment = one ds_load_b64.
// ---------------------------------------------------------------------------
template <int C, int N, bool GATE, bool FINAL>
__global__ __launch_bounds__(384) void conv_block(
    const float* __restrict__ in, const float* __restrict__ Kt,
    const float* __restrict__ bias, const float* __restrict__ la,
    const float* __restrict__ ab, const float* __restrict__ as,
    float* __restrict__ out, float* __restrict__ ld_ps) {
    constexpr int MT   = (C + 15) / 16;
    constexpr int CPAD = MT * 16;
    constexpr int NT   = N / 16;
    constexpr int COLP = N + 3;            // padded halo row
    constexpr int CPRE = C / 4;
    constexpr int N2   = 2 * N;
    constexpr int NTH  = MT * NT * 32;
    constexpr int C4   = C / 4;
    constexpr int KS   = 9 * C4;           // total K steps

    // channel-pair interleaved: element (ci, r, cc) at ((ci/2)*3+r)*COLP*2 + cc*2 + (ci&1)
    __shared__ float s_in[C * 3 * COLP];
    __shared__ float s_scale[C];
    __shared__ float s_bias[C];

    const int tid = threadIdx.x;
    const int b = blockIdx.x / N;
    const int y = blockIdx.x % N;

    for (int c = tid; c < C; c += NTH) {
        s_scale[c] = expf(as[c]);
        s_bias[c]  = ab[c];
    }
    __syncthreads();

    // stage squeezed + actnormed input: 3 wrapped rows, N+2 wrapped cols, C channels
    const float* inb = in + (size_t)b * CPRE * N2 * N2;
    for (int e = tid; e < C * 3 * (N + 2); e += NTH) {
        int ci = e / (3 * (N + 2));
        int r  = (e / (N + 2)) % 3;
        int cc = e % (N + 2);
        int yy = (y + r - 1 + N) % N;
        int xx = (cc - 1 + N) % N;
        int cpre = ci >> 2, s1 = (ci >> 1) & 1, s2 = ci & 1;
        float v = inb[((size_t)cpre * N2 + (2 * yy + s1)) * N2 + (2 * xx + s2)];
        s_in[(((ci >> 1) * 3 + r) * COLP + cc) * 2 + (ci & 1)] = v * s_scale[ci] + s_bias[ci];
    }
    __syncthreads();

    const int wave = tid >> 5, lane = tid & 31;
    const int mt = wave / NT, nt = wave % NT;
    const int co0 = mt * 16, xb = nt * 16;
    const int np = lane & 15, kh = lane >> 4;   // f32 frag: lanes<16 k={0,1}, lanes>=16 k={2,3}

    // per-lane constant address parts
    const float* aptr = Kt + ((size_t)kh * CPAD + co0 + np) * 2;   // + s*4*CPAD per step
    const int bl = (3 * kh * COLP + xb + np) * 2;                  // lane part of LDS addr

    auto loadFrag = [&](int s, v2f& a, v2f& bf) {
        int tap = s / C4;
        int c4  = s - tap * C4;
        int dy  = tap / 3;
        int dx  = tap - dy * 3;
        a = *(const v2f*)(aptr + (size_t)s * 4 * CPAD);
        int baddr = ((6 * c4 + dy) * COLP + dx) * 2 + bl;   // pair (kk, kk+1) adjacent
        bf = *(const v2f*)&s_in[baddr];
    };

    v8f acc = {};
    v2f a0, b0, a1, b1;
    loadFrag(0, a0, b0);
    loadFrag(KS > 1 ? 1 : 0, a1, b1);
    for (int s = 0; s < KS; ++s) {
        int s2 = s + 2 < KS ? s + 2 : KS - 1;   // clamped prefetch (no OOB)
        v2f an, bn;
        loadFrag(s2, an, bn);
        acc = __builtin_amdgcn_wmma_f32_16x16x4_f32(
            false, a0, false, b0, (short)0, acc, false, false);
        a0 = a1; b0 = b1;
        a1 = an; b1 = bn;
    }

    // epilogue: D layout — VGPR j: lanes0-15 M=j, lanes16-31 M=j+8; N=lane&15
    float ldl = 0.f;
    const int x = xb + np;
    for (int j = 0; j < 8; ++j) {
        int co = co0 + j + 8 * kh;
        if ((C % 16 != 0) && co >= C) continue;   // padded rows (block 1 only)
        float v = acc[j] + bias[((size_t)co * N + y) * N + x];
        float* op = out + (((size_t)b * C + co) * N + y) * N + x;
        if (GATE) {
            float alpha = expf(la[co]);
            float sgn = (v < 0.f) ? -1.f : (v > 0.f ? 1.f : 0.f);
            float yv = sgn / alpha * log1pf(alpha * fabsf(v));
            ldl += -alpha * fabsf(yv);
            *op = yv;
        } else if (FINAL) {
            ldl += -0.5f * v * v - HALF_LOG_2PI;
            *op = v;
        } else {
            *op = v;
        }
    }
    for (int off = 16; off > 0; off >>= 1) ldl += __shfl_down(ldl, off);
    if (lane == 0) atomicAdd(&ld_ps[b], ldl);
}

// ---------------------------------------------------------------------------
// spectral conv logdet: per frequency (u,v) build 9-tap DFT matrix (c x c
// complex) in LDS, complex LU with partial pivoting, sum log|pivots|
// ---------------------------------------------------------------------------
template <int C, int N>
__global__ __launch_bounds__(256) void conv_logdet(const float* __restrict__ K,
                                                   float* __restrict__ ld_sc) {
    extern __shared__ float M[];   // C*C complex, interleaved re/im
    __shared__ float rv[256];
    __shared__ int   ri[256];
    const int t = threadIdx.x;
    const int u = blockIdx.x / N, v = blockIdx.x % N;

    float pr[9], pi[9];
    const float w = -6.283185307179586f / (float)N;
    for (int tap = 0; tap < 9; ++tap) {
        int dy = tap / 3 - 1, dx = tap % 3 - 1;
        float ang = w * (float)(u * dy + v * dx);
        pr[tap] = cosf(ang);
        pi[tap] = sinf(ang);
    }
    for (int e = t; e < C * C; e += 256) {
        const float* kp = K + (size_t)e * 9;
        float re = 0.f, im = 0.f;
        for (int tap = 0; tap < 9; ++tap) { float kv = kp[tap]; re += kv * pr[tap]; im += kv * pi[tap]; }
        M[2 * e] = re;
        M[2 * e + 1] = im;
    }
    __syncthreads();

    for (int k = 0; k < C - 1; ++k) {
        float best = -1.f; int bi = k;
        for (int r = k + t; r < C; r += 256) {
            float m = fabsf(M[2 * (r * C + k)]) + fabsf(M[2 * (r * C + k) + 1]);
            if (m > best) { best = m; bi = r; }
        }
        rv[t] = best; ri[t] = bi;
        __syncthreads();
        for (int s = 128; s > 0; s >>= 1) {
            if (t < s && rv[t + s] > rv[t]) { rv[t] = rv[t + s]; ri[t] = ri[t + s]; }
            __syncthreads();
        }
        int piv = ri[0];
        if (piv != k) {
            for (int j = k + t; j < C; j += 256) {
                float tr = M[2 * (k * C + j)], ti = M[2 * (k * C + j) + 1];
                M[2 * (k * C + j)]     = M[2 * (piv * C + j)];
                M[2 * (k * C + j) + 1] = M[2 * (piv * C + j) + 1];
                M[2 * (piv * C + j)]     = tr;
                M[2 * (piv * C + j) + 1] = ti;
            }
        }
        __syncthreads();
        float prr = M[2 * (k * C + k)], pii = M[2 * (k * C + k) + 1];
        float inv = 1.f / (prr * prr + pii * pii);
        for (int i = k + 1 + t; i < C; i += 256) {
            float ar = M[2 * (i * C + k)], ai = M[2 * (i * C + k) + 1];
            float fr = (ar * prr + ai * pii) * inv;
            float fi = (ai * prr - ar * pii) * inv;
            for (int j = k + 1; j < C; ++j) {
                float br = M[2 * (k * C + j)], bim = M[2 * (k * C + j) + 1];
                M[2 * (i * C + j)]     -= fr * br - fi * bim;
                M[2 * (i * C + j) + 1] -= fr * bim + fi * br;
            }
        }
        __syncthreads();
    }
    float s = 0.f;
    for (int d = t; d < C; d += 256) {
        float dr = M[2 * (d * C + d)], di = M[2 * (d * C + d) + 1];
        s += 0.5f * logf(dr * dr + di * di);
    }
    rv[t] = s;
    __syncthreads();
    for (int st = 128; st > 0; st >>= 1) {
        if (t < st) rv[t] += rv[t + st];
        __syncthreads();
    }
    if (t == 0) atomicAdd(ld_sc, rv[0]);
}

// ---------------------------------------------------------------------------
// actnorm scalar logdet: sum_i n_i^2 * sum(as_i)
// ---------------------------------------------------------------------------
__global__ void actnorm_ld(const float* __restrict__ as1, const float* __restrict__ as2,
                           const float* __restrict__ as3, float* __restrict__ ld_sc) {
    __shared__ float red[256];
    int t = threadIdx.x;
    float s = 0.f;
    if (t < 12) s += 4096.f * as1[t];
    if (t < 48) s += 1024.f * as2[t];
    if (t < 192) s += 256.f * as3[t];
    red[t] = s;
    __syncthreads();
    for (int st = 128; st > 0; st >>= 1) {
        if (t < st) red[t] += red[t + st];
        __syncthreads();
    }
    if (t == 0) atomicAdd(ld_sc, red[0]);
}

__global__ void finalize(const float* __restrict__ ld_ps, const float* __restrict__ ld_sc,
                         float* __restrict__ logp) {
    int t = threadIdx.x;
    if (t < 128) logp[t] = ld_ps[t] + *ld_sc;
}

// ---------------------------------------------------------------------------
extern "C" void kernel_launch(void* const* d_in, const int* in_sizes, int n_in,
                              void* d_out, int out_size, void* d_ws, size_t ws_size,
                              hipStream_t stream) {
    (void)in_sizes; (void)n_in; (void)out_size; (void)ws_size;
    const float* x   = (const float*)d_in[0];
    const float* k1  = (const float*)d_in[1];
    const float* b1  = (const float*)d_in[2];
    const float* la1 = (const float*)d_in[3];
    const float* ab1 = (const float*)d_in[4];
    const float* as1 = (const float*)d_in[5];
    const float* k2  = (const float*)d_in[6];
    const float* b2  = (const float*)d_in[7];
    const float* la2 = (const float*)d_in[8];
    const float* ab2 = (const float*)d_in[9];
    const float* as2 = (const float*)d_in[10];
    const float* k3  = (const float*)d_in[11];
    const float* b3  = (const float*)d_in[12];
    const float* la3 = (const float*)d_in[13];
    const float* ab3 = (const float*)d_in[14];
    const float* as3 = (const float*)d_in[15];
    float* out = (float*)d_out;

    float* ws    = (float*)d_ws;
    float* ld_ps = ws;                 // [128]
    float* ld_sc = ws + 128;           // [1]
    float* bufA  = ws + 256;           // ping
    float* bufB  = bufA + ACT_ELEMS;   // pong
    float* kt1   = bufB + ACT_ELEMS;   // 9*12*16  = 1728 (CPAD=16)
    float* kt2   = kt1 + 1728;         // 9*48*48  = 20736
    float* kt3   = kt2 + 20736;        // 9*192*192 = 331776

    const float c0 = 49152.0f * (logf(0.1f) + logf(0.99f));

    init_accum<<<1, 256, 0, stream>>>(ld_ps, ld_sc, c0);
    transpose_w<<<8, 256, 0, stream>>>(k1, kt1, 12, 16);
    transpose_w<<<96, 256, 0, stream>>>(k2, kt2, 48, 48);
    transpose_w<<<1296, 256, 0, stream>>>(k3, kt3, 192, 192);

    logit_fwd<<<dim3(192, 128), 256, 0, stream>>>(x, bufA, ld_ps);

    conv_block<12, 64, true, false><<<128 * 64, 128, 0, stream>>>(
        bufA, kt1, b1, la1, ab1, as1, bufB, ld_ps);
    conv_block<48, 32, true, false><<<128 * 32, 192, 0, stream>>>(
        bufB, kt2, b2, la2, ab2, as2, bufA, ld_ps);
    conv_block<192, 16, false, true><<<128 * 16, 384, 0, stream>>>(
        bufA, kt3, b3, la3, ab3, as3, out, ld_ps);

    conv_logdet<12, 64><<<64 * 64, 256, 12 * 12 * 2 * sizeof(float), stream>>>(k1, ld_sc);
    conv_logdet<48, 32><<<32 * 32, 256, 48 * 48 * 2 * sizeof(float), stream>>>(k2, ld_sc);
    conv_logdet<192, 16><<<16 * 16, 256, 192 * 192 * 2 * sizeof(float), stream>>>(k3, ld_sc);

    actnorm_ld<<<1, 256, 0, stream>>>(as1, as2, as3, ld_sc);
    finalize<<<1, 128, 0, stream>>>(ld_ps, ld_sc, out + ACT_ELEMS);
}